// craft_mse_loss_36180804502178
// MI455X (gfx1250) — compile-verified
//
#include <hip/hip_runtime.h>
#include <hip/hip_bf16.h>
#include <stdint.h>
#include <stddef.h>

// ---------------------------------------------------------------------------
// CRAFT MSE loss with per-sample OHEM (exact 3-level radix select on float
// bits) for MI455X / gfx1250.  Bandwidth-bound: streams inputs through LDS
// using CDNA5 async global->LDS B128 copies (512B/wave/instr) with
// ASYNCcnt-based double buffering.
// ---------------------------------------------------------------------------

#define Bn 32
#define HW (384 * 384)                 // 147456 elements per sample
#define BLOCKS_PER_SAMPLE 36
#define THREADS 256
#define VEC 4                          // floats per lane per async B128
#define TILE (THREADS * VEC)           // 1024 elements per tile
#define EPB (HW / BLOCKS_PER_SAMPLE)   // 4096 elements per block
#define ITERS (EPB / TILE)             // 4 tiles per block
#define NB_HI 2048                     // 11-bit radix levels 0,1
#define NB_LO 1024                     // 10-bit radix level 2
#define NPART (Bn * BLOCKS_PER_SAMPLE) // 1152 partial sums
#define EPSLN 1e-7f

struct Ws {
  unsigned int hist[Bn][NB_HI];  // reused across radix levels
  unsigned int fgcnt[Bn];
  unsigned int bgcnt[Bn];
  unsigned int prefix[Bn];       // selected high bits so far
  int          kleft[Bn];        // remaining rank within selected bin
  float        thresh[Bn];       // exact k-th largest bg loss
  float        pnum[NPART];
  float        pden[NPART];
};
static_assert(sizeof(Ws) < 300 * 1024, "workspace footprint");

// ---------------- CDNA5 async global->LDS helpers --------------------------

typedef int v4i __attribute__((vector_size(16)));
typedef __attribute__((address_space(1))) int   as1_i32;
typedef __attribute__((address_space(3))) int   as3_i32;
typedef __attribute__((address_space(1))) v4i   as1_v4i;
typedef __attribute__((address_space(3))) v4i   as3_v4i;
typedef __attribute__((address_space(3))) float as3_f32;

#if __has_builtin(__builtin_amdgcn_global_load_async_to_lds_b128)
#define HAVE_ASYNC_B128 1
#endif
#if __has_builtin(__builtin_amdgcn_global_load_async_to_lds_b32)
#define HAVE_ASYNC_B32 1
#endif
#if __has_builtin(__builtin_amdgcn_s_wait_asynccnt)
#define HAVE_WAIT_ASYNC 1
#endif

// One 16B copy per lane: 512B per wave per instruction, tracked by ASYNCcnt.
__device__ __forceinline__ void async_b128(const float* g, float* l) {
#if defined(HAVE_ASYNC_B128)
  __builtin_amdgcn_global_load_async_to_lds_b128((as1_v4i*)g, (as3_v4i*)l, 0, 0);
#elif defined(HAVE_ASYNC_B32)
  __builtin_amdgcn_global_load_async_to_lds_b32((as1_i32*)(g + 0), (as3_i32*)(l + 0), 0, 0);
  __builtin_amdgcn_global_load_async_to_lds_b32((as1_i32*)(g + 1), (as3_i32*)(l + 1), 0, 0);
  __builtin_amdgcn_global_load_async_to_lds_b32((as1_i32*)(g + 2), (as3_i32*)(l + 2), 0, 0);
  __builtin_amdgcn_global_load_async_to_lds_b32((as1_i32*)(g + 3), (as3_i32*)(l + 3), 0, 0);
#else
  unsigned lofs = (unsigned)(size_t)(as3_f32*)l;
  asm volatile("global_load_async_to_lds_b128 %0, %1, off"
               :: "v"(lofs), "v"(g) : "memory");
#endif
}

// async instructions issued per 7-array tile (for the double-buffer wait)
#if defined(HAVE_ASYNC_B128) || !defined(HAVE_ASYNC_B32)
#define ASYNCS_PER_TILE 7
#else
#define ASYNCS_PER_TILE 28
#endif

#if defined(HAVE_WAIT_ASYNC)
#define WAIT_ASYNC(N)                                   \
  do {                                                  \
    __builtin_amdgcn_s_wait_asynccnt(N);                \
    asm volatile("" ::: "memory");                      \
  } while (0)
#else
#define WAIT_ASYNC(N) asm volatile("s_wait_asynccnt %0" :: "i"(N) : "memory")
#endif

// ---------------- loss (matches reference op order, round-to-nearest) ------

__device__ __forceinline__ float loss_of(float rtv, float atv, float rpv,
                                         float apv, float cv) {
  float dr = __fadd_rn(rtv, -rpv);
  float da = __fadd_rn(atv, -apv);
  float lr = __fmul_rn(__fmul_rn(dr, dr), cv);
  float la = __fmul_rn(__fmul_rn(da, da), cv);
  return __fadd_rn(lr, la);
}

__device__ __forceinline__ float4 ld4(const float* p) {
  return *(const float4*)p;   // ds_load_b128 from LDS
}

// ---------------- kernels ---------------------------------------------------

__global__ void k_init(Ws* __restrict__ ws) {
  unsigned i = blockIdx.x * blockDim.x + threadIdx.x;
  unsigned* h = &ws->hist[0][0];
  if (i < (unsigned)(Bn * NB_HI)) {
    h[i] = 0u;
  } else {
    unsigned j = i - (unsigned)(Bn * NB_HI);
    if (j < Bn)           ws->fgcnt[j] = 0u;
    else if (j < 2u * Bn) ws->bgcnt[j - Bn] = 0u;
  }
}

// Radix-histogram pass. LEVEL 0: also counts fg/bg. Inputs are streamed
// through LDS with double-buffered async B128 copies.
template <int LEVEL>
__global__ void k_hist(const float* __restrict__ rt, const float* __restrict__ at,
                       const float* __restrict__ rp, const float* __restrict__ ap,
                       const float* __restrict__ cf, const float* __restrict__ fg,
                       const float* __restrict__ bg, Ws* __restrict__ ws) {
  __shared__ unsigned int lh[NB_HI];
  __shared__ __align__(16) float stage[2][7][TILE];
  const int tid = threadIdx.x;
  const int b   = blockIdx.x / BLOCKS_PER_SAMPLE;
  const int blk = blockIdx.x % BLOCKS_PER_SAMPLE;
  const int NB  = (LEVEL == 2) ? NB_LO : NB_HI;
  const int lo  = tid * VEC;

  for (int i = tid; i < NB; i += THREADS) lh[i] = 0u;
  __syncthreads();

  unsigned pfx = 0u;
  if (LEVEL > 0) pfx = ws->prefix[b];

  const long long base = (long long)b * HW + (long long)blk * EPB;
  unsigned fgc = 0, bgc = 0;

  // prime buffer 0
  {
    const long long i = base + lo;
    async_b128(rt + i, &stage[0][0][lo]);
    async_b128(at + i, &stage[0][1][lo]);
    async_b128(rp + i, &stage[0][2][lo]);
    async_b128(ap + i, &stage[0][3][lo]);
    async_b128(cf + i, &stage[0][4][lo]);
    async_b128(fg + i, &stage[0][5][lo]);
    async_b128(bg + i, &stage[0][6][lo]);
  }
  for (int it = 0; it < ITERS; ++it) {
    if (it + 1 < ITERS) {
      const int nb = (it + 1) & 1;
      const long long i = base + (long long)(it + 1) * TILE + lo;
      async_b128(rt + i, &stage[nb][0][lo]);
      async_b128(at + i, &stage[nb][1][lo]);
      async_b128(rp + i, &stage[nb][2][lo]);
      async_b128(ap + i, &stage[nb][3][lo]);
      async_b128(cf + i, &stage[nb][4][lo]);
      async_b128(fg + i, &stage[nb][5][lo]);
      async_b128(bg + i, &stage[nb][6][lo]);
      WAIT_ASYNC(ASYNCS_PER_TILE);  // newest batch in flight; previous done
    } else {
      WAIT_ASYNC(0);
    }
    const int bf = it & 1;
    const float4 v_rt = ld4(&stage[bf][0][lo]);
    const float4 v_at = ld4(&stage[bf][1][lo]);
    const float4 v_rp = ld4(&stage[bf][2][lo]);
    const float4 v_ap = ld4(&stage[bf][3][lo]);
    const float4 v_cf = ld4(&stage[bf][4][lo]);
    const float4 v_fg = ld4(&stage[bf][5][lo]);
    const float4 v_bg = ld4(&stage[bf][6][lo]);
    const float rts[VEC] = {v_rt.x, v_rt.y, v_rt.z, v_rt.w};
    const float ats[VEC] = {v_at.x, v_at.y, v_at.z, v_at.w};
    const float rps[VEC] = {v_rp.x, v_rp.y, v_rp.z, v_rp.w};
    const float aps[VEC] = {v_ap.x, v_ap.y, v_ap.z, v_ap.w};
    const float cfs[VEC] = {v_cf.x, v_cf.y, v_cf.z, v_cf.w};
    const float fgs[VEC] = {v_fg.x, v_fg.y, v_fg.z, v_fg.w};
    const float bgs[VEC] = {v_bg.x, v_bg.y, v_bg.z, v_bg.w};
#pragma unroll
    for (int j = 0; j < VEC; ++j) {
      const float l = loss_of(rts[j], ats[j], rps[j], aps[j], cfs[j]);
      const unsigned bits = __float_as_uint(l);  // l >= 0 -> bits monotonic
      if (LEVEL == 0) {
        if (fgs[j] > 0.5f) fgc++;
        if (bgs[j] > 0.5f) { bgc++; atomicAdd(&lh[bits >> 21], 1u); }
      } else if (LEVEL == 1) {
        if (bgs[j] > 0.5f && (bits >> 21) == pfx)
          atomicAdd(&lh[(bits >> 10) & 0x7FFu], 1u);
      } else {
        if (bgs[j] > 0.5f && (bits >> 10) == pfx)
          atomicAdd(&lh[bits & 0x3FFu], 1u);
      }
    }
  }
  if (LEVEL == 0) {
    if (fgc) atomicAdd(&ws->fgcnt[b], fgc);
    if (bgc) atomicAdd(&ws->bgcnt[b], bgc);
  }
  __syncthreads();
  for (int i = tid; i < NB; i += THREADS)
    if (lh[i]) atomicAdd(&ws->hist[b][i], lh[i]);
}

// One block per sample: descending scan of the histogram to locate the bin
// containing the k-th largest value; also zeroes the histogram for the next
// radix level. Integer-only -> deterministic and exact.
template <int LEVEL>
__global__ void k_select(Ws* __restrict__ ws) {
  __shared__ unsigned int lh[NB_HI];
  const int b  = blockIdx.x;
  const int NB = (LEVEL == 2) ? NB_LO : NB_HI;
  for (int i = threadIdx.x; i < NB; i += blockDim.x) {
    lh[i] = ws->hist[b][i];
    ws->hist[b][i] = 0u;
  }
  __syncthreads();
  if (threadIdx.x == 0) {
    long long k;
    if (LEVEL == 0) {
      const long long fgn = ws->fgcnt[b];
      const long long bgn = ws->bgcnt[b];
      k = 3 * fgn;                 // (fg_num * 3.0) exact for counts < 2^24
      if (k < 10000) k = 10000;
      if (k > bgn)   k = bgn;      // min(bg_num, neg_num)
    } else {
      k = ws->kleft[b];
    }
    if (k <= 0) {                  // no background pixels: hard mask is empty
      ws->kleft[b] = 0;
      if (LEVEL == 0) ws->prefix[b] = 0u;
      if (LEVEL == 2) ws->thresh[b] = __uint_as_float(0x7F800000u); // +inf
    } else {
      int sel = -1;
      for (int bin = NB - 1; bin >= 0; --bin) {
        const long long c = (long long)lh[bin];
        if (k <= c) { sel = bin; break; }
        k -= c;
      }
      if (sel < 0) { sel = 0; k = 1; }   // defensive clamp (unreachable)
      if (LEVEL == 0) {
        ws->prefix[b] = (unsigned)sel;
        ws->kleft[b]  = (int)k;
      } else if (LEVEL == 1) {
        ws->prefix[b] = (ws->prefix[b] << 11) | (unsigned)sel;
        ws->kleft[b]  = (int)k;
      } else {
        const unsigned bits = (ws->prefix[b] << 10) | (unsigned)sel;
        ws->thresh[b] = __uint_as_float(bits);  // exact k-th largest bg loss
      }
    }
  }
}

// Final masked reduction: num = sum(l * train), den = sum(conf * train).
__global__ void k_final(const float* __restrict__ rt, const float* __restrict__ at,
                        const float* __restrict__ rp, const float* __restrict__ ap,
                        const float* __restrict__ cf, const float* __restrict__ fg,
                        const float* __restrict__ bg, Ws* __restrict__ ws) {
  __shared__ __align__(16) float stage[2][7][TILE];
  __shared__ float sn[THREADS];
  __shared__ float sd[THREADS];
  const int tid = threadIdx.x;
  const int b   = blockIdx.x / BLOCKS_PER_SAMPLE;
  const int blk = blockIdx.x % BLOCKS_PER_SAMPLE;
  const int lo  = tid * VEC;
  const float th = ws->thresh[b];
  const long long base = (long long)b * HW + (long long)blk * EPB;

  float num = 0.0f, den = 0.0f;
  {
    const long long i = base + lo;
    async_b128(rt + i, &stage[0][0][lo]);
    async_b128(at + i, &stage[0][1][lo]);
    async_b128(rp + i, &stage[0][2][lo]);
    async_b128(ap + i, &stage[0][3][lo]);
    async_b128(cf + i, &stage[0][4][lo]);
    async_b128(fg + i, &stage[0][5][lo]);
    async_b128(bg + i, &stage[0][6][lo]);
  }
  for (int it = 0; it < ITERS; ++it) {
    if (it + 1 < ITERS) {
      const int nb = (it + 1) & 1;
      const long long i = base + (long long)(it + 1) * TILE + lo;
      async_b128(rt + i, &stage[nb][0][lo]);
      async_b128(at + i, &stage[nb][1][lo]);
      async_b128(rp + i, &stage[nb][2][lo]);
      async_b128(ap + i, &stage[nb][3][lo]);
      async_b128(cf + i, &stage[nb][4][lo]);
      async_b128(fg + i, &stage[nb][5][lo]);
      async_b128(bg + i, &stage[nb][6][lo]);
      WAIT_ASYNC(ASYNCS_PER_TILE);
    } else {
      WAIT_ASYNC(0);
    }
    const int bf = it & 1;
    const float4 v_rt = ld4(&stage[bf][0][lo]);
    const float4 v_at = ld4(&stage[bf][1][lo]);
    const float4 v_rp = ld4(&stage[bf][2][lo]);
    const float4 v_ap = ld4(&stage[bf][3][lo]);
    const float4 v_cf = ld4(&stage[bf][4][lo]);
    const float4 v_fg = ld4(&stage[bf][5][lo]);
    const float4 v_bg = ld4(&stage[bf][6][lo]);
    const float rts[VEC] = {v_rt.x, v_rt.y, v_rt.z, v_rt.w};
    const float ats[VEC] = {v_at.x, v_at.y, v_at.z, v_at.w};
    const float rps[VEC] = {v_rp.x, v_rp.y, v_rp.z, v_rp.w};
    const float aps[VEC] = {v_ap.x, v_ap.y, v_ap.z, v_ap.w};
    const float cfs[VEC] = {v_cf.x, v_cf.y, v_cf.z, v_cf.w};
    const float fgs[VEC] = {v_fg.x, v_fg.y, v_fg.z, v_fg.w};
    const float bgs[VEC] = {v_bg.x, v_bg.y, v_bg.z, v_bg.w};
#pragma unroll
    for (int j = 0; j < VEC; ++j) {
      const float l = loss_of(rts[j], ats[j], rps[j], aps[j], cfs[j]);
      const bool tr = (fgs[j] > 0.5f) || ((bgs[j] > 0.5f) && (l >= th));
      if (tr) { num += l; den += cfs[j]; }
    }
  }
  // deterministic fixed-shape block reduction
  sn[tid] = num; sd[tid] = den;
  __syncthreads();
  for (int s = THREADS / 2; s > 0; s >>= 1) {
    if (tid < s) { sn[tid] += sn[tid + s]; sd[tid] += sd[tid + s]; }
    __syncthreads();
  }
  if (tid == 0) { ws->pnum[blockIdx.x] = sn[0]; ws->pden[blockIdx.x] = sd[0]; }
}

__global__ void k_finalize(Ws* __restrict__ ws, float* __restrict__ out) {
  __shared__ float sn[THREADS];
  __shared__ float sd[THREADS];
  const int tid = threadIdx.x;
  float n = 0.0f, d = 0.0f;
  for (int i = tid; i < NPART; i += THREADS) {  // fixed order per thread
    n += ws->pnum[i];
    d += ws->pden[i];
  }
  sn[tid] = n; sd[tid] = d;
  __syncthreads();
  for (int s = THREADS / 2; s > 0; s >>= 1) {
    if (tid < s) { sn[tid] += sn[tid + s]; sd[tid] += sd[tid + s]; }
    __syncthreads();
  }
  if (tid == 0) out[0] = sn[0] / (sd[0] + EPSLN);
}

// ---------------------------------------------------------------------------

extern "C" void kernel_launch(void* const* d_in, const int* in_sizes, int n_in,
                              void* d_out, int out_size, void* d_ws, size_t ws_size,
                              hipStream_t stream) {
  (void)in_sizes; (void)n_in; (void)out_size; (void)ws_size;
  const float* rt = (const float*)d_in[0];  // region_true
  const float* at = (const float*)d_in[1];  // affinity_true
  const float* rp = (const float*)d_in[2];  // region_pred
  const float* ap = (const float*)d_in[3];  // affinity_pred
  const float* cf = (const float*)d_in[4];  // confidence
  const float* fg = (const float*)d_in[5];  // fg_mask
  const float* bg = (const float*)d_in[6];  // bg_mask
  float* out = (float*)d_out;
  Ws* ws = (Ws*)d_ws;

  const int init_items = Bn * NB_HI + 2 * Bn;
  k_init<<<(init_items + THREADS - 1) / THREADS, THREADS, 0, stream>>>(ws);

  const dim3 grid(Bn * BLOCKS_PER_SAMPLE);
  k_hist<0><<<grid, THREADS, 0, stream>>>(rt, at, rp, ap, cf, fg, bg, ws);
  k_select<0><<<Bn, THREADS, 0, stream>>>(ws);
  k_hist<1><<<grid, THREADS, 0, stream>>>(rt, at, rp, ap, cf, fg, bg, ws);
  k_select<1><<<Bn, THREADS, 0, stream>>>(ws);
  k_hist<2><<<grid, THREADS, 0, stream>>>(rt, at, rp, ap, cf, fg, bg, ws);
  k_select<2><<<Bn, THREADS, 0, stream>>>(ws);
  k_final<<<grid, THREADS, 0, stream>>>(rt, at, rp, ap, cf, fg, bg, ws);
  k_finalize<<<1, THREADS, 0, stream>>>(ws, out);
}